// TropicalMultiHeadAttention_23295902613833
// MI455X (gfx1250) — compile-verified
//
#include <hip/hip_runtime.h>
#include <hip/hip_bf16.h>

#define BATCH   2
#define LSEQ    1024
#define DMODEL  1024
#define HEADS   16
#define DKH     64

typedef float v2f __attribute__((ext_vector_type(2)));
typedef float v8f __attribute__((ext_vector_type(8)));

__device__ __forceinline__ v8f wmma4(v2f a, v2f b, v8f c) {
    // D = A(16x4, f32) * B(4x16, f32) + C(16x16, f32)  -- full f32 matrix op
    return __builtin_amdgcn_wmma_f32_16x16x4_f32(false, a, false, b, (short)0, c,
                                                 false, false);
}

// ---------------------------------------------------------------------------
// Y[m][n] = sum_k X[m][k] * W[n][k] + bias[n]
// Wave tile: 32(M) x 64(N). Block = 4 waves -> 32 x 256.  Grid: (N/256, M/32).
// Register double-buffered: fragment set for k-chunk c+1 loads while the 32
// WMMAs of chunk c execute (hides L2 latency at 1 wave/SIMD occupancy).
// ---------------------------------------------------------------------------
__global__ __launch_bounds__(128) void gemm_xwT_kernel(
    const float* __restrict__ X, const float* __restrict__ W,
    const float* __restrict__ bias, float* __restrict__ Y,
    int M, int N, int Kd)
{
    const int tid  = threadIdx.x;
    const int wave = tid >> 5;
    const int lane = tid & 31;
    const int half = lane >> 4;
    const int l15  = lane & 15;

    const int m0 = blockIdx.y * 32;
    const int n0 = blockIdx.x * 256 + wave * 64;
    const int koff = 2 * half;

    // loop-invariant per-lane element offsets (32-bit); uniform k advances in SALU
    const unsigned xo0 = (unsigned)(m0 + l15) * (unsigned)Kd + (unsigned)koff;
    const unsigned xo1 = xo0 + 16u * (unsigned)Kd;
    unsigned wo[4];
    #pragma unroll
    for (int t = 0; t < 4; ++t)
        wo[t] = (unsigned)(n0 + 16 * t + l15) * (unsigned)Kd + (unsigned)koff;

    v8f acc[2][4] = {};
    v2f A0[2][4], B0[4][4], A1[2][4], B1[4][4];

    auto ldset = [&](v2f (&Af)[2][4], v2f (&Bf)[4][4], int k) {
        #pragma unroll
        for (int s = 0; s < 4; ++s) {
            Af[0][s] = *(const v2f*)(X + xo0 + k + 4 * s);
            Af[1][s] = *(const v2f*)(X + xo1 + k + 4 * s);
            #pragma unroll
            for (int t = 0; t < 4; ++t)
                Bf[t][s] = *(const v2f*)(W + wo[t] + k + 4 * s);
        }
    };
    auto domm = [&](v2f (&Af)[2][4], v2f (&Bf)[4][4]) {
        #pragma unroll
        for (int s = 0; s < 4; ++s)
            #pragma unroll
            for (int t = 0; t < 4; ++t) {
                acc[0][t] = wmma4(Af[0][s], Bf[t][s], acc[0][t]);
                acc[1][t] = wmma4(Af[1][s], Bf[t][s], acc[1][t]);
            }
    };

    ldset(A0, B0, 0);
    #pragma unroll 1
    for (int k = 0; k + 32 <= Kd; k += 32) {
        ldset(A1, B1, k + 16);      // in flight during domm(A0,B0)
        domm(A0, B0);
        if (k + 32 < Kd)
            ldset(A0, B0, k + 32);  // in flight during domm(A1,B1)
        domm(A1, B1);
    }

    #pragma unroll
    for (int t = 0; t < 4; ++t) {
        const int n = n0 + 16 * t + l15;
        const float bv = bias[n];
        #pragma unroll
        for (int r = 0; r < 8; ++r) {
            Y[(size_t)(m0 + r + 8 * half) * N + n]      = acc[0][t][r] + bv;
            Y[(size_t)(m0 + 16 + r + 8 * half) * N + n] = acc[1][t][r] + bv;
        }
    }
}

// ---------------------------------------------------------------------------
// G[tok][h] = sigmoid(dot(query[tok], Wg[h]) + bg[h]).  Block = 16 waves,
// one wave per head; grid = B*L tokens.
// ---------------------------------------------------------------------------
__global__ __launch_bounds__(512) void gate_kernel(
    const float* __restrict__ query, const float* __restrict__ Wg,
    const float* __restrict__ bg, float* __restrict__ G)
{
    const int tok  = blockIdx.x;
    const int h    = threadIdx.x >> 5;
    const int lane = threadIdx.x & 31;

    const float* q = query + (size_t)tok * DMODEL;
    const float* w = Wg + (size_t)h * DMODEL;
    float acc = 0.0f;
    for (int i = lane; i < DMODEL; i += 32) acc = fmaf(q[i], w[i], acc);
    #pragma unroll
    for (int m = 16; m >= 1; m >>= 1) acc += __shfl_xor(acc, m, 32);
    if (lane == 0)
        G[(size_t)tok * HEADS + h] = 1.0f / (1.0f + __expf(-(acc + bg[h])));
}

// ---------------------------------------------------------------------------
// Flash attention with tropical/classical blended scores.
// Block = 4 waves, 64 q rows per (b,h); wave owns a 16-row q tile.
// Grid: (L/64, H, B).  K/V tiles double-buffered through registers so the
// global loads for block kb+1 overlap the whole compute phase of block kb.
// ---------------------------------------------------------------------------
__global__ __launch_bounds__(128) void attn_kernel(
    const float* __restrict__ Q, const float* __restrict__ K,
    const float* __restrict__ V, const float* __restrict__ G,
    const float* __restrict__ log_temps, float* __restrict__ AO)
{
    const int b  = blockIdx.z;
    const int h  = blockIdx.y;
    const int q0 = blockIdx.x * 64;

    const int tid  = threadIdx.x;
    const int wave = tid >> 5;
    const int lane = tid & 31;
    const int half = lane >> 4;
    const int l15  = lane & 15;

    __shared__ float Qs[64][68];       // 64 q rows x DK, padded
    __shared__ float Ks[16][68];       // k-block tile
    __shared__ float Vs[16][68];
    __shared__ float Ps[4][16][18];    // per-wave P transpose scratch

    // ---- stage Q tile (each thread: 32 floats) ----
    {
        const int row = tid >> 1;
        const int c0  = (tid & 1) * 32;
        const float* src = Q + ((size_t)(b * LSEQ + q0 + row)) * DMODEL + h * DKH + c0;
        #pragma unroll
        for (int i = 0; i < 32; i += 4)
            *(float4*)&Qs[row][c0 + i] = *(const float4*)(src + i);
    }

    // per-head temperature: tau = clip(exp(lt), 0.02, 10); scores *= SCALE/tau
    float tau = __expf(log_temps[h]);
    tau = fminf(fmaxf(tau, 0.02f), 10.0f);
    const float sfac = 0.125f / tau;   // SCALE = DK^-0.5 = 1/8

    // per-lane gate for its 8 C-layout rows
    float g8[8];
    #pragma unroll
    for (int r = 0; r < 8; ++r) {
        const int tok = q0 + 16 * wave + r + 8 * half;
        g8[r] = G[((size_t)(b * LSEQ + tok)) * HEADS + h];
    }

    float Macc[8], Ssum[8];
    #pragma unroll
    for (int r = 0; r < 8; ++r) { Macc[r] = -__builtin_inff(); Ssum[r] = 0.0f; }
    v8f Oacc[4] = {};

    // K/V register staging (8 floats of K, 8 of V per thread)
    const int srow = tid >> 3;
    const int sc0  = (tid & 7) * 8;
    float4 nk0, nk1, nv0, nv1;
    auto ldkv = [&](int kb2) {
        const size_t base =
            ((size_t)(b * LSEQ + kb2 * 16 + srow)) * DMODEL + h * DKH + sc0;
        nk0 = *(const float4*)(K + base);
        nk1 = *(const float4*)(K + base + 4);
        nv0 = *(const float4*)(V + base);
        nv1 = *(const float4*)(V + base + 4);
    };
    ldkv(0);

    __syncthreads();

    #pragma unroll 1
    for (int kb = 0; kb < LSEQ / 16; ++kb) {
        __syncthreads();               // previous block's readers done
        *(float4*)&Ks[srow][sc0]     = nk0;
        *(float4*)&Ks[srow][sc0 + 4] = nk1;
        *(float4*)&Vs[srow][sc0]     = nv0;
        *(float4*)&Vs[srow][sc0 + 4] = nv1;
        if (kb + 1 < LSEQ / 16) ldkv(kb + 1);   // overlaps compute below
        __syncthreads();

        // K column for this lane's C-layout column (broadcast reads across halves)
        float kcol[64];
        #pragma unroll
        for (int d = 0; d < 64; d += 4) {
            float4 kv = *(const float4*)&Ks[l15][d];
            kcol[d] = kv.x; kcol[d + 1] = kv.y; kcol[d + 2] = kv.z; kcol[d + 3] = kv.w;
        }

        // ---- classical scores: S = Qtile(16x64) x K^T(64x16), 16 f32 WMMAs ----
        v8f sc = {};
        #pragma unroll
        for (int t = 0; t < 16; ++t) {
            const int dd = 4 * t + 2 * half;
            v2f a; a.x = Qs[16 * wave + l15][dd]; a.y = Qs[16 * wave + l15][dd + 1];
            v2f bb; bb.x = kcol[dd]; bb.y = kcol[dd + 1];
            sc = wmma4(a, bb, sc);
        }

        // ---- tropical max-plus + gate blend (VALU; kcol in regs, Q via LDS bcast) ----
        float s8[8];
        #pragma unroll
        for (int r = 0; r < 8; ++r) {
            const int mrow = 16 * wave + r + 8 * half;
            float tr = -__builtin_inff();
            #pragma unroll
            for (int d = 0; d < 64; d += 4) {
                float4 qv = *(const float4*)&Qs[mrow][d];
                tr = fmaxf(tr, qv.x + kcol[d]);
                tr = fmaxf(tr, qv.y + kcol[d + 1]);
                tr = fmaxf(tr, qv.z + kcol[d + 2]);
                tr = fmaxf(tr, qv.w + kcol[d + 3]);
            }
            const float g = g8[r];
            s8[r] = (g * tr + (1.0f - g) * sc[r]) * sfac;
        }

        // ---- online softmax (row reductions across each 16-lane half) ----
        float p8[8], rs8[8];
        #pragma unroll
        for (int r = 0; r < 8; ++r) {
            float mx = s8[r];
            mx = fmaxf(mx, __shfl_xor(mx, 1, 16));
            mx = fmaxf(mx, __shfl_xor(mx, 2, 16));
            mx = fmaxf(mx, __shfl_xor(mx, 4, 16));
            mx = fmaxf(mx, __shfl_xor(mx, 8, 16));
            const float mnew = fmaxf(Macc[r], mx);
            const float resc = __expf(Macc[r] - mnew);
            const float p    = __expf(s8[r] - mnew);
            float ps = p;
            ps += __shfl_xor(ps, 1, 16);
            ps += __shfl_xor(ps, 2, 16);
            ps += __shfl_xor(ps, 4, 16);
            ps += __shfl_xor(ps, 8, 16);
            Ssum[r] = Ssum[r] * resc + ps;
            Macc[r] = mnew;
            p8[r] = p; rs8[r] = resc;
        }
        #pragma unroll
        for (int c = 0; c < 4; ++c)
            #pragma unroll
            for (int r = 0; r < 8; ++r)
                Oacc[c][r] *= rs8[r];

        // ---- P transpose through per-wave LDS, then O += P(16x16) x V(16x64) ----
        #pragma unroll
        for (int r = 0; r < 8; ++r)
            Ps[wave][r + 8 * half][l15] = p8[r];
        asm volatile("s_wait_dscnt 0" ::: "memory");

        #pragma unroll
        for (int t = 0; t < 4; ++t) {
            const int kk = 4 * t + 2 * half;
            v2f a; a.x = Ps[wave][l15][kk]; a.y = Ps[wave][l15][kk + 1];
            #pragma unroll
            for (int c = 0; c < 4; ++c) {
                v2f bb; bb.x = Vs[kk][16 * c + l15]; bb.y = Vs[kk + 1][16 * c + l15];
                Oacc[c] = wmma4(a, bb, Oacc[c]);
            }
        }
    }

    // ---- normalize and write attention output in (B,L,D) layout ----
    #pragma unroll
    for (int r = 0; r < 8; ++r) {
        const float inv = 1.0f / Ssum[r];
        const int tok = q0 + 16 * wave + r + 8 * half;
        float* dst = AO + ((size_t)(b * LSEQ + tok)) * DMODEL + h * DKH;
        #pragma unroll
        for (int c = 0; c < 4; ++c)
            dst[16 * c + l15] = Oacc[c][r] * inv;
    }
}

// ---------------------------------------------------------------------------
extern "C" void kernel_launch(void* const* d_in, const int* in_sizes, int n_in,
                              void* d_out, int out_size, void* d_ws, size_t ws_size,
                              hipStream_t stream) {
    (void)in_sizes; (void)n_in; (void)out_size; (void)ws_size;
    const float* query = (const float*)d_in[0];
    const float* key   = (const float*)d_in[1];
    const float* value = (const float*)d_in[2];
    const float* Wq    = (const float*)d_in[3];
    const float* bq    = (const float*)d_in[4];
    const float* Wk    = (const float*)d_in[5];
    const float* bk    = (const float*)d_in[6];
    const float* Wv    = (const float*)d_in[7];
    const float* bv    = (const float*)d_in[8];
    const float* Wo    = (const float*)d_in[9];
    const float* bo    = (const float*)d_in[10];
    const float* Wg    = (const float*)d_in[11];
    const float* bg    = (const float*)d_in[12];
    const float* ltmp  = (const float*)d_in[13];

    float* ws = (float*)d_ws;
    const size_t MT = (size_t)BATCH * LSEQ;          // 2048 tokens
    float* Qws  = ws;
    float* Kws  = Qws + MT * DMODEL;
    float* Vws  = Kws + MT * DMODEL;
    float* AOws = Vws + MT * DMODEL;
    float* Gws  = AOws + MT * DMODEL;

    const dim3 gblk(128);
    const dim3 ggrd(DMODEL / 256, (unsigned)(MT / 32));

    gemm_xwT_kernel<<<ggrd, gblk, 0, stream>>>(query, Wq, bq, Qws, (int)MT, DMODEL, DMODEL);
    gemm_xwT_kernel<<<ggrd, gblk, 0, stream>>>(key,   Wk, bk, Kws, (int)MT, DMODEL, DMODEL);
    gemm_xwT_kernel<<<ggrd, gblk, 0, stream>>>(value, Wv, bv, Vws, (int)MT, DMODEL, DMODEL);
    gate_kernel<<<dim3((unsigned)MT), dim3(512), 0, stream>>>(query, Wg, bg, Gws);
    attn_kernel<<<dim3(LSEQ / 64, HEADS, BATCH), dim3(128), 0, stream>>>(
        Qws, Kws, Vws, Gws, ltmp, AOws);
    gemm_xwT_kernel<<<ggrd, gblk, 0, stream>>>(AOws, Wo, bo, (float*)d_out, (int)MT, DMODEL, DMODEL);
}